// NeRFRenderer_6210522710171
// MI455X (gfx1250) — compile-verified
//
#include <hip/hip_runtime.h>

// NeRF sample_pdf (inverse-CDF importance sampling) for MI455X / gfx1250.
// Memory-bound (~1.07 GB @ 23.3 TB/s => ~46us floor). Cumsum done with
// full-precision f32 WMMA (V_WMMA_F32_16X16X4_F32): prefix-sum == matmul
// with a triangular ones matrix, 16 rays per wave. Bins tile staged to LDS
// via gfx1250 async global->LDS loads (ASYNCcnt).

typedef __attribute__((ext_vector_type(2))) float v2f;
typedef __attribute__((ext_vector_type(8))) float v8f;
typedef __attribute__((ext_vector_type(4))) int   v4i;

// Address-space-qualified element types for the async builtin's pointer params.
typedef __attribute__((address_space(1))) v4i as1_v4i;   // global (__device__)
typedef __attribute__((address_space(3))) v4i as3_v4i;   // LDS (__shared__)

#define N_BINS        128
#define N_W           127   // weights per ray
#define N_SAMP        128
#define RAYS_PER_WAVE 16
#define WAVES_PER_BLK 2
#define THREADS       (WAVES_PER_BLK * 32)
#define EPS           1e-5f

// Async copy of one 16-byte element per lane from global to LDS.
__device__ __forceinline__ void async_copy_b128(const void* gsrc, void* ldst) {
#if __has_builtin(__builtin_amdgcn_global_load_async_to_lds_b128)
    // gfx1250 intrinsic: (as1 int4* src, as3 int4* dst, imm offset, imm cpol)
    __builtin_amdgcn_global_load_async_to_lds_b128(
        (as1_v4i*)(uintptr_t)gsrc,
        (as3_v4i*)(unsigned)(uintptr_t)ldst,   // low 32 bits of flat LDS addr == LDS offset
        0, 0);
#else
    unsigned lds_off = (unsigned)(uintptr_t)ldst;
    unsigned long long gaddr = (unsigned long long)(uintptr_t)gsrc;
    asm volatile("global_load_async_to_lds_b128 %0, %1, off"
                 :: "v"(lds_off), "v"(gaddr) : "memory");
#endif
}

__device__ __forceinline__ void wait_asynccnt0() {
#if __has_builtin(__builtin_amdgcn_s_wait_asynccnt)
    __builtin_amdgcn_s_wait_asynccnt(0);
#else
    asm volatile("s_wait_asynccnt 0" ::: "memory");
#endif
}

__global__ __launch_bounds__(THREADS)
void nerf_sample_pdf_kernel(const float* __restrict__ bins,
                            const float* __restrict__ weights,
                            const float* __restrict__ u,
                            float* __restrict__ out,
                            int n_rays)
{
    // Per-wave LDS regions: raw CDF (pos 0..127, pos0 = 0) and bins tile.
    __shared__ __align__(16) float lds_cdf [WAVES_PER_BLK][RAYS_PER_WAVE][N_BINS];
    __shared__ __align__(16) float lds_bins[WAVES_PER_BLK][RAYS_PER_WAVE][N_BINS];

    const int lane  = threadIdx.x & 31;
    const int wave  = threadIdx.x >> 5;
    const int tile  = blockIdx.x * WAVES_PER_BLK + wave;
    const int rbase = tile * RAYS_PER_WAVE;
    if (rbase >= n_rays) return;   // uniform per-wave; never triggers for full tiles

    // ------------- Phase 0: bins tile -> LDS (async b128, ASYNCcnt) ----------
    {
        const float4* src = reinterpret_cast<const float4*>(bins + (size_t)rbase * N_BINS);
        float4*       dst = reinterpret_cast<float4*>(&lds_bins[wave][0][0]);
        #pragma unroll
        for (int i = 0; i < (RAYS_PER_WAVE * N_BINS / 4) / 32; ++i)   // 16 iters
            async_copy_b128(&src[i * 32 + lane], &dst[i * 32 + lane]);
    }

    // ---------------- Phase 1: cumsum(w+EPS) via f32 WMMA --------------------
    // A (16x4 f32): lane L holds row M = L%16; VGPR0/1 hold K = k0+2*(L/16), +1.
    // B (4x16)    : synthesized triangular ones, mirrored layout (chunk-invariant).
    // D (16x16)   : vgpr v, lane L -> row (v + 8*(L/16)), col (L%16).
    const int rowSel = lane >> 4;     // 0 for lanes 0-15, 1 for lanes 16-31
    const int mrow   = lane & 15;     // A-operand ray within tile
    const int ncol   = lane & 15;     // B/D column within chunk

    const float* wrow = weights + (size_t)(rbase + mrow) * N_W;

    // B is identical for every chunk: precompute the 4 K-step operands once.
    v2f bmat[4];
    #pragma unroll
    for (int ks = 0; ks < 4; ++ks) {
        const int kk = ks * 4 + 2 * rowSel;           // chunk-local K pair
        bmat[ks].x = (kk     <= ncol) ? 1.0f : 0.0f;  // U[k][n] = (k <= n)
        bmat[ks].y = (kk + 1 <= ncol) ? 1.0f : 0.0f;
    }

    v8f acc;                          // running per-ray offsets (D layout, col-replicated)
    #pragma unroll
    for (int i = 0; i < 8; ++i) acc[i] = 0.0f;

    #pragma unroll
    for (int c = 0; c < 8; ++c) {
        v8f d;
        #pragma unroll
        for (int i = 0; i < 8; ++i) d[i] = 0.0f;

        #pragma unroll
        for (int ks = 0; ks < 4; ++ks) {
            const int k0 = c * 16 + ks * 4 + 2 * rowSel;  // this lane's K pair
            v2f a;
            a.x = (k0     < N_W) ? (wrow[k0]     + EPS) : 0.0f;
            a.y = (k0 + 1 < N_W) ? (wrow[k0 + 1] + EPS) : 0.0f;
            d = __builtin_amdgcn_wmma_f32_16x16x4_f32(
                    /*neg_a=*/false, a, /*neg_b=*/false, bmat[ks],
                    /*c_mod=*/(short)0, d, /*reuse_a=*/false, /*reuse_b=*/false);
        }

        // cdf[idx+1] = within-chunk prefix + carried offset; drop idx==127.
        #pragma unroll
        for (int v = 0; v < 8; ++v) {
            const int ray = v + 8 * rowSel;
            const int idx = c * 16 + ncol;
            const float val = d[v] + acc[v];
            if (idx < N_BINS - 1)
                lds_cdf[wave][ray][idx + 1] = val;
        }
        // carry: broadcast chunk total (col 15) within each 16-lane half.
        #pragma unroll
        for (int v = 0; v < 8; ++v)
            acc[v] += __shfl(d[v], 15, 16);
    }
    if (lane < RAYS_PER_WAVE) lds_cdf[wave][lane][0] = 0.0f;

    wait_asynccnt0();     // bins tile resident in LDS
    __syncthreads();

    // ---------------- Phase 2: per-sample search + lerp ----------------------
    const float* urow0 = u   + (size_t)rbase * N_SAMP;
    float*       orow0 = out + (size_t)rbase * N_SAMP;

    #pragma unroll 1
    for (int r = 0; r < RAYS_PER_WAVE; ++r) {
        const float* crow = &lds_cdf [wave][r][0];
        const float* brow = &lds_bins[wave][r][0];
        const float  total = crow[N_BINS - 1];        // sum(w + EPS)
        const float  inv_t = 1.0f / total;

        #pragma unroll
        for (int s4 = 0; s4 < N_SAMP / 32; ++s4) {
            const int   s   = s4 * 32 + lane;
            const float uu  = urow0[(size_t)r * N_SAMP + s];
            const float key = uu * total;             // search raw cdf vs u*total

            // searchsorted(crow[0..127], key, side='right'), branchless.
            int pos = 0;
            #pragma unroll
            for (int step = 64; step > 0; step >>= 1) {
                const int np = pos + step;
                pos = (crow[np - 1] <= key) ? np : pos;
            }
            const int below = (pos - 1 > 0) ? pos - 1 : 0;
            const int above = (pos < N_BINS - 1) ? pos : N_BINS - 1;

            const float cb = crow[below] * inv_t;
            const float ca = crow[above] * inv_t;
            const float bb = brow[below];
            const float ba = brow[above];

            float denom = ca - cb;
            denom = (denom < EPS) ? 1.0f : denom;
            const float t = (uu - cb) / denom;
            orow0[(size_t)r * N_SAMP + s] = bb + t * (ba - bb);
        }
    }
}

extern "C" void kernel_launch(void* const* d_in, const int* in_sizes, int n_in,
                              void* d_out, int out_size, void* d_ws, size_t ws_size,
                              hipStream_t stream) {
    const float* bins    = (const float*)d_in[0];
    const float* weights = (const float*)d_in[1];
    const float* u       = (const float*)d_in[2];
    float*       out     = (float*)d_out;

    const int n_rays = in_sizes[0] / N_BINS;                 // 524288
    const int tiles  = (n_rays + RAYS_PER_WAVE - 1) / RAYS_PER_WAVE;
    const int blocks = (tiles + WAVES_PER_BLK - 1) / WAVES_PER_BLK;

    nerf_sample_pdf_kernel<<<blocks, THREADS, 0, stream>>>(bins, weights, u, out, n_rays);
}